// GATSCL_87316685127963
// MI455X (gfx1250) — compile-verified
//
#include <hip/hip_runtime.h>
#include <math.h>

typedef __attribute__((ext_vector_type(2))) float v2f;
typedef __attribute__((ext_vector_type(8))) float v8f;

// ---------------- order-preserving float<->uint for atomic max ----------------
__device__ __forceinline__ unsigned f2u_ord(float f) {
    unsigned b = __float_as_uint(f);
    return (b & 0x80000000u) ? ~b : (b | 0x80000000u);
}
__device__ __forceinline__ float u2f_ord(unsigned u) {
    unsigned b = (u & 0x80000000u) ? (u & 0x7FFFFFFFu) : ~u;
    return __uint_as_float(b);
}
__device__ __forceinline__ float lrelu(float v) { return v > 0.0f ? v : 0.2f * v; }

// ---------------- fp32 WMMA GEMM: out[n x 128] = A[n x 128] @ W[128 x 128] (+bias) ----
// 256 threads = 8 wave32s; wave w computes columns [16w, 16w+16) of a 16-row block.
// V_WMMA_F32_16X16X4_F32 lane layout (ISA 7.12.2):
//   A (16x4):  lanes 0-15: M=lane, VGPR0=K0,VGPR1=K1 ; lanes 16-31: M=lane-16, K2/K3
//   B (4x16):  VGPR0: lanes0-15 row K0, lanes16-31 row K2 ; VGPR1: K1 / K3
//   C (16x16): VGPR v: lanes0-15 -> M=v, lanes16-31 -> M=v+8 ; N=lane%16
__global__ __launch_bounds__(256) void gemm128_wmma(const float* __restrict__ A,
                                                    const float* __restrict__ W,
                                                    const float* __restrict__ bias,
                                                    float* __restrict__ out, int n) {
    __shared__ float As[16][128];
    const int row0 = blockIdx.x * 16;
    const int t = threadIdx.x;
    const bool full = (row0 + 16) <= n;

    if (full) {
        // fast path: 16x128 tile = 512 float4; 2 b128 transfers per thread
        const float4* Ag = (const float4*)(A + (size_t)row0 * 128);
        float4* As4 = (float4*)&As[0][0];
        As4[t] = Ag[t];
        As4[t + 256] = Ag[t + 256];
    } else {
        for (int i = 0; i < 8; ++i) {
            int off = t + i * 256;
            int r = off >> 7, c = off & 127;
            int gr = row0 + r;
            As[r][c] = (gr < n) ? A[(size_t)gr * 128 + c] : 0.0f;
        }
    }
    __syncthreads();

    const int wave = t >> 5;
    const int lane = t & 31;
    const int half = lane >> 4;   // 0 or 1
    const int m = lane & 15;
    const int j = wave * 16 + m;  // output column (also B-fragment column)

    v8f acc = {};
#pragma unroll 4
    for (int k0 = 0; k0 < 128; k0 += 4) {
        const int kb = k0 + half * 2;
        v2f a, b;
        a.x = As[m][kb];
        a.y = As[m][kb + 1];
        b.x = W[kb * 128 + j];
        b.y = W[(kb + 1) * 128 + j];
        acc = __builtin_amdgcn_wmma_f32_16x16x4_f32(false, a, false, b,
                                                    (short)0, acc, false, false);
    }

    const float bj = bias ? bias[j] : 0.0f;
    if (full) {
        // unconditional store tail: 8 global_store_b32, clause-friendly
        float* op = out + ((size_t)row0 + half * 8) * 128 + j;
#pragma unroll
        for (int v = 0; v < 8; ++v) op[(size_t)v * 128] = acc[v] + bj;
    } else {
#pragma unroll
        for (int v = 0; v < 8; ++v) {
            int gr = row0 + v + half * 8;
            if (gr < n) out[(size_t)gr * 128 + j] = acc[v] + bj;
        }
    }
}

// ---------------- per-(node,head) attention coefficients ----------------
__global__ __launch_bounds__(256) void attn_kernel(const float* __restrict__ xp,
                                                   const float* __restrict__ att_src,
                                                   const float* __restrict__ att_dst,
                                                   float* __restrict__ a_src,
                                                   float* __restrict__ a_dst, int n) {
    int idx = blockIdx.x * blockDim.x + threadIdx.x;
    if (idx >= n * 8) return;
    int nd = idx >> 3, h = idx & 7;
    const float4* xv = (const float4*)(xp + (size_t)nd * 128 + h * 16);
    const float4* sv = (const float4*)(att_src + h * 16);
    const float4* dv = (const float4*)(att_dst + h * 16);
    float ss = 0.0f, dd = 0.0f;
#pragma unroll
    for (int q = 0; q < 4; ++q) {
        float4 xq = xv[q], aq = sv[q], bq = dv[q];
        ss += xq.x * aq.x + xq.y * aq.y + xq.z * aq.z + xq.w * aq.w;
        dd += xq.x * bq.x + xq.y * bq.y + xq.z * bq.z + xq.w * bq.w;
    }
    a_src[idx] = ss;
    a_dst[idx] = dd;
}

// ---------------- init: z <- bias broadcast, e_max <- -inf, e_sum <- 0 --------
__global__ __launch_bounds__(256) void init_kernel(const float* __restrict__ b_gat,
                                                   float* __restrict__ z,
                                                   unsigned* __restrict__ emax,
                                                   float* __restrict__ esum, int n) {
    int idx = blockIdx.x * blockDim.x + threadIdx.x;
    if (idx >= n * 128) return;
    z[idx] = b_gat[idx & 127];
    if (idx < n * 8) {
        emax[idx] = f2u_ord(-INFINITY);
        esum[idx] = 0.0f;
    }
}

// ---------------- edge pass 1: segment max of leaky-relu logits ----------------
__global__ __launch_bounds__(256) void edge_max_kernel(const int* __restrict__ src,
                                                       const int* __restrict__ dst,
                                                       const float* __restrict__ a_src,
                                                       const float* __restrict__ a_dst,
                                                       unsigned* __restrict__ emax,
                                                       int E, int EE) {
    int e = blockIdx.x * blockDim.x + threadIdx.x;
    if (e >= EE) return;
    int s, d;
    if (e < E) { s = src[e]; d = dst[e]; } else { s = d = e - E; }
#pragma unroll
    for (int h = 0; h < 8; ++h) {
        float v = lrelu(a_src[s * 8 + h] + a_dst[d * 8 + h]);
        atomicMax(&emax[d * 8 + h], f2u_ord(v));
    }
}

// ---------------- edge pass 2: segment sum of exp(logit - max) ----------------
__global__ __launch_bounds__(256) void edge_sum_kernel(const int* __restrict__ src,
                                                       const int* __restrict__ dst,
                                                       const float* __restrict__ a_src,
                                                       const float* __restrict__ a_dst,
                                                       const unsigned* __restrict__ emax,
                                                       float* __restrict__ esum,
                                                       int E, int EE) {
    int e = blockIdx.x * blockDim.x + threadIdx.x;
    if (e >= EE) return;
    int s, d;
    if (e < E) { s = src[e]; d = dst[e]; } else { s = d = e - E; }
#pragma unroll
    for (int h = 0; h < 8; ++h) {
        float v = lrelu(a_src[s * 8 + h] + a_dst[d * 8 + h]);
        float p = expf(v - u2f_ord(emax[d * 8 + h]));
        atomicAdd(&esum[d * 8 + h], p);
    }
}

// ---------------- edge pass 3: weighted scatter (one thread per edge-head) -----
__global__ __launch_bounds__(256) void edge_scatter_kernel(const int* __restrict__ src,
                                                           const int* __restrict__ dst,
                                                           const float* __restrict__ a_src,
                                                           const float* __restrict__ a_dst,
                                                           const unsigned* __restrict__ emax,
                                                           const float* __restrict__ esum,
                                                           const float* __restrict__ xp,
                                                           float* __restrict__ z,
                                                           int E, int EE) {
    int idx = blockIdx.x * blockDim.x + threadIdx.x;
    if (idx >= EE * 8) return;
    int e = idx >> 3, h = idx & 7;
    int s, d;
    if (e < E) { s = src[e]; d = dst[e]; } else { s = d = e - E; }
    float v = lrelu(a_src[s * 8 + h] + a_dst[d * 8 + h]);
    float alpha = expf(v - u2f_ord(emax[d * 8 + h])) / (esum[d * 8 + h] + 1e-16f);
    const float4* xs = (const float4*)(xp + (size_t)s * 128 + h * 16);
    float* zd = z + (size_t)d * 128 + h * 16;
#pragma unroll
    for (int q = 0; q < 4; ++q) {
        float4 m = xs[q];  // L2-resident gather (xp = 51 MB < 192 MB L2)
        atomicAdd(zd + q * 4 + 0, m.x * alpha);
        atomicAdd(zd + q * 4 + 1, m.y * alpha);
        atomicAdd(zd + q * 4 + 2, m.z * alpha);
        atomicAdd(zd + q * 4 + 3, m.w * alpha);
    }
}

extern "C" void kernel_launch(void* const* d_in, const int* in_sizes, int n_in,
                              void* d_out, int out_size, void* d_ws, size_t ws_size,
                              hipStream_t stream) {
    const float* x       = (const float*)d_in[0];
    const int*   ei      = (const int*)d_in[1];
    const float* W_gat   = (const float*)d_in[2];
    const float* att_src = (const float*)d_in[3];
    const float* att_dst = (const float*)d_in[4];
    const float* b_gat   = (const float*)d_in[5];
    const float* W_fc1   = (const float*)d_in[6];
    const float* b_fc1   = (const float*)d_in[7];
    const float* W_fc2   = (const float*)d_in[8];
    const float* b_fc2   = (const float*)d_in[9];

    const int n  = in_sizes[0] / 128;
    const int E  = in_sizes[1] / 2;
    const int EE = E + n;  // edges + self loops

    float* z  = (float*)d_out;                       // [n,128]
    float* z1 = z + (size_t)n * 128;                 // [n,128]
    float* z2 = z1 + (size_t)n * 128;                // [n,128]
    float* xp = z2;                                  // stage xp in z2 slot (dead before final GEMM)

    float*    a_src = (float*)d_ws;                  // [n,8]
    float*    a_dst = a_src + (size_t)n * 8;         // [n,8]
    unsigned* emax  = (unsigned*)(a_dst + (size_t)n * 8);  // [n,8]
    float*    esum  = (float*)(emax + (size_t)n * 8);      // [n,8]

    const int* srcI = ei;
    const int* dstI = ei + E;

    const dim3 blk(256);
    const int gemmBlocks = (n + 15) / 16;

    // 1) xp = x @ W_gat  (fp32 WMMA)
    gemm128_wmma<<<gemmBlocks, blk, 0, stream>>>(x, W_gat, nullptr, xp, n);
    // 2) attention coefficients
    attn_kernel<<<(n * 8 + 255) / 256, blk, 0, stream>>>(xp, att_src, att_dst, a_src, a_dst, n);
    // 3) init accumulators (z <- b_gat, emax <- -inf, esum <- 0)
    init_kernel<<<(n * 128 + 255) / 256, blk, 0, stream>>>(b_gat, z, emax, esum, n);
    // 4-6) segment softmax + weighted aggregation over edges
    edge_max_kernel<<<(EE + 255) / 256, blk, 0, stream>>>(srcI, dstI, a_src, a_dst, emax, E, EE);
    edge_sum_kernel<<<(EE + 255) / 256, blk, 0, stream>>>(srcI, dstI, a_src, a_dst, emax, esum, E, EE);
    edge_scatter_kernel<<<(EE * 8 + 255) / 256, blk, 0, stream>>>(srcI, dstI, a_src, a_dst,
                                                                  emax, esum, xp, z, E, EE);
    // 7-8) FC heads (fp32 WMMA); z2 GEMM overwrites the xp staging area (now dead)
    gemm128_wmma<<<gemmBlocks, blk, 0, stream>>>(z, W_fc1, b_fc1, z1, n);
    gemm128_wmma<<<gemmBlocks, blk, 0, stream>>>(z, W_fc2, b_fc2, z2, n);
}